// st_VQEmbedding_52243982188938
// MI455X (gfx1250) — compile-verified
//
#include <hip/hip_runtime.h>

typedef __attribute__((ext_vector_type(2))) float v2f;
typedef __attribute__((ext_vector_type(8))) float v8f;

#define VQ_D 64
#define LDS_STRIDE 68   // 64 + 4 floats pad -> conflict-free ds_load_b64 pattern

// ---- kernel 0: halfe2[k] = 0.5 * ||embedding[k]||^2 ------------------------
__global__ void vq_halfnorm_kernel(const float* __restrict__ emb,
                                   float* __restrict__ halfe2, int K) {
    int k = blockIdx.x * blockDim.x + threadIdx.x;
    if (k >= K) return;
    const float4* w = (const float4*)(emb + (size_t)k * VQ_D);
    float s = 0.f;
#pragma unroll
    for (int i = 0; i < VQ_D / 4; ++i) {
        float4 v = w[i];
        s += v.x * v.x + v.y * v.y + v.z * v.z + v.w * v.w;
    }
    halfe2[k] = 0.5f * s;
}

// ---- kernel 1: async-stage codebook to LDS, WMMA scores, argmin, gather ----
__global__ __launch_bounds__(1024) void vq_assign_gather_kernel(
    const float* __restrict__ x, const float* __restrict__ emb,
    const float* __restrict__ halfe2, float* __restrict__ out,
    int BN, int K)
{
    extern __shared__ float smem[];
    float* sW = smem;                      // K rows x LDS_STRIDE floats (padded)
    float* sH = smem + (size_t)K * LDS_STRIDE;  // K floats of 0.5*||w||^2

    // ---- stage codebook + halfe2 into LDS with CDNA5 async loads ----------
    {
        const unsigned wbase = (unsigned)(size_t)sW;
        const unsigned hbase = (unsigned)(size_t)sH;
        const unsigned long long egbl = (unsigned long long)emb;
        const unsigned long long hgbl = (unsigned long long)halfe2;
        const int nchunks = K * (VQ_D / 4);          // 16B chunks of codebook
        for (int i = threadIdx.x; i < nchunks; i += blockDim.x) {
            const int row = i >> 4, c = i & 15;      // 16 chunks per row
            unsigned laddr = wbase + (unsigned)(row * (LDS_STRIDE * 4) + c * 16);
            unsigned goff  = (unsigned)(i * 16);
            asm volatile("global_load_async_to_lds_b128 %0, %1, %2"
                         :: "v"(laddr), "v"(goff), "s"(egbl) : "memory");
        }
        for (int i = threadIdx.x; i < K / 4; i += blockDim.x) {
            unsigned laddr = hbase + (unsigned)(i * 16);
            unsigned goff  = (unsigned)(i * 16);
            asm volatile("global_load_async_to_lds_b128 %0, %1, %2"
                         :: "v"(laddr), "v"(goff), "s"(hgbl) : "memory");
        }
        asm volatile("s_wait_asynccnt 0" ::: "memory");
    }
    __syncthreads();

    const int lane = threadIdx.x & 31;
    const int tile = blockIdx.x * (blockDim.x >> 5) + (threadIdx.x >> 5);
    if (tile * 16 < BN) {                      // wave-uniform: EXEC all-1s inside
        const int row_base = tile * 16;
        const int col      = lane & 15;        // code / output column stripe
        const int dsel     = (lane >> 4) << 1; // K-offset 0 (lanes 0-15) or 2

        // A fragment: 16 k-steps of 16x16x4 covering D=64 (ISA 16x4 f32 layout)
        v2f a[16];
        const float* xr = x + (size_t)(row_base + col) * VQ_D + dsel;
#pragma unroll
        for (int kk = 0; kk < 16; ++kk)
            a[kk] = *(const v2f*)(xr + 4 * kk);

        float best[8];
        int   bidx[8];
#pragma unroll
        for (int j = 0; j < 8; ++j) { best[j] = -3.402823466e38f; bidx[j] = 0; }

        // Sweep codebook from LDS, two 16-code tiles per pass (dual accumulators)
        for (int kt = 0; kt < K; kt += 32) {
            const int code0 = kt + col;
            const int code1 = kt + 16 + col;
            const float* w0 = sW + code0 * LDS_STRIDE + dsel;
            const float* w1 = sW + code1 * LDS_STRIDE + dsel;
            v8f c0 = {};
            v8f c1 = {};
#pragma unroll
            for (int kk = 0; kk < 16; ++kk) {
                v2f b0 = *(const v2f*)(w0 + 4 * kk);
                v2f b1 = *(const v2f*)(w1 + 4 * kk);
                c0 = __builtin_amdgcn_wmma_f32_16x16x4_f32(
                    false, a[kk], false, b0, (short)0, c0, false, false);
                c1 = __builtin_amdgcn_wmma_f32_16x16x4_f32(
                    false, a[kk], false, b1, (short)0, c1, false, false);
            }
            const float h0 = sH[code0];
            const float h1 = sH[code1];
#pragma unroll
            for (int j = 0; j < 8; ++j) {
                float v0 = c0[j] - h0;             // row g*8+j, code0
                if (v0 > best[j]) { best[j] = v0; bidx[j] = code0; }
                float v1 = c1[j] - h1;
                if (v1 > best[j]) { best[j] = v1; bidx[j] = code1; }
            }
        }

        // Cross-lane argmax within each 16-lane half (codes are lane-striped);
        // xor masks 1..8 stay inside each half on wave32. Tie -> lower index.
#pragma unroll
        for (int j = 0; j < 8; ++j) {
#pragma unroll
            for (int m = 1; m <= 8; m <<= 1) {
                float ov = __shfl_xor(best[j], m, 32);
                int   oi = __shfl_xor(bidx[j], m, 32);
                if (ov > best[j] || (ov == best[j] && oi < bidx[j])) {
                    best[j] = ov; bidx[j] = oi;
                }
            }
        }

        // Gather from LDS copy: half-wave g owns rows g*8..g*8+7;
        // 16 lanes x float4 = one fully-coalesced 256B output row.
        const int g = lane >> 4;
#pragma unroll
        for (int j = 0; j < 8; ++j) {
            const int row  = row_base + g * 8 + j;
            const int code = bidx[j];
            const float* src = sW + code * LDS_STRIDE + col * 4;
            float4 v = make_float4(src[0], src[1], src[2], src[3]);
            ((float4*)(out + (size_t)row * VQ_D))[col] = v;
        }
    }
}

extern "C" void kernel_launch(void* const* d_in, const int* in_sizes, int n_in,
                              void* d_out, int out_size, void* d_ws, size_t ws_size,
                              hipStream_t stream) {
    const float* x   = (const float*)d_in[0];   // z_e_x  [B,N,D] f32
    const float* emb = (const float*)d_in[1];   // embedding [K,D] f32
    float*       out = (float*)d_out;           // z_q_x  [B,N,D] f32
    float*    halfe2 = (float*)d_ws;            // K floats of scratch

    const int BN = in_sizes[0] / VQ_D;          // 65536
    const int K  = in_sizes[1] / VQ_D;          // 512

    vq_halfnorm_kernel<<<(K + 255) / 256, 256, 0, stream>>>(emb, halfe2, K);

    const int threads       = 1024;             // 32 waves amortize the LDS stage
    const int waves_per_blk = threads / 32;
    const int tiles         = BN / 16;          // one wave per 16-row tile
    const int blocks        = (tiles + waves_per_blk - 1) / waves_per_blk;
    const size_t shmem = (size_t)K * LDS_STRIDE * sizeof(float)   // padded codebook
                       + (size_t)K * sizeof(float);               // halfe2
    vq_assign_gather_kernel<<<blocks, threads, shmem, stream>>>(
        x, emb, halfe2, out, BN, K);
}